// HeterogeneousGNN_23055384445847
// MI455X (gfx1250) — compile-verified
//
#include <hip/hip_runtime.h>

typedef __attribute__((ext_vector_type(2))) float v2f;
typedef __attribute__((ext_vector_type(8))) float v8f;

#define NNODES 20000
#define HDIM 256

// ---------------------------------------------------------------------------
// C[N,256] = A[N,K] @ W[K,256] (+ bias), fp32 via V_WMMA_F32_16X16X4_F32.
// grid.x = N/32 row tiles; 256 threads = 8 waves; wave w -> cols [16w,16w+16).
// Each wave computes TWO 16x16 output tiles (rows row0..+15 and row0+16..+31),
// reusing each B fragment for 2 WMMAs.
// A tile (32 x K) staged via async direct-to-LDS copies (ASYNCcnt path),
// padded LDS row stride 260 -> conflict-free column reads.
// Fragment layouts (per ISA 7.12.2, 32-bit data):
//   A 16x4 : lanes 0-15: v0=K+0,v1=K+1 ; lanes 16-31: v0=K+2,v1=K+3 ; M=lane%16
//   B 4x16 : lanes 0-15 rows K+0/K+1, lanes 16-31 rows K+2/K+3 ; N=lane%16
//   D 16x16: vgpr i -> row i (lanes 0-15) / row i+8 (lanes 16-31)
// ---------------------------------------------------------------------------
__global__ __launch_bounds__(256) void gemm_bias_wmma(
    const float* __restrict__ A, const float* __restrict__ W,
    const float* __restrict__ bias, float* __restrict__ C, int K)
{
    __shared__ float lds_a[32 * 260];
    const int tid  = threadIdx.x;
    const int wave = tid >> 5;
    const int lane = tid & 31;
    const int row0 = blockIdx.x << 5;   // 32 rows per block

    // ---- async copy A tile [32 x K] global -> LDS, 16B per issue ----
    const float* Ablk = A + (size_t)row0 * K;
    const int nvec = (32 * K) >> 2;                    // float4 chunks
    for (int i = tid; i < nvec; i += 256) {
        int e = i << 2;
        int r = e / K;
        int c = e - r * K;
        unsigned lds_off = (unsigned)(uintptr_t)(&lds_a[r * 260 + c]);
        unsigned g_off   = (unsigned)((r * K + c) << 2);   // byte offset
        asm volatile("global_load_async_to_lds_b128 %0, %1, %2 offset:0"
                     :: "v"(lds_off), "v"(g_off), "s"(Ablk) : "memory");
    }
    asm volatile("s_wait_asynccnt 0x0" ::: "memory");
    __syncthreads();

    const int m   = lane & 15;
    const int col = (wave << 4) + m;
    const int kh  = (lane >> 4) << 1;   // 0 for lanes 0-15, 2 for lanes 16-31

    v8f acc0 = {0.f, 0.f, 0.f, 0.f, 0.f, 0.f, 0.f, 0.f};
    v8f acc1 = {0.f, 0.f, 0.f, 0.f, 0.f, 0.f, 0.f, 0.f};
    #pragma unroll 8
    for (int k = 0; k < K; k += 4) {
        v2f b;
        b.x = W[(size_t)(k + kh) * HDIM + col];
        b.y = W[(size_t)(k + kh + 1) * HDIM + col];
        float2 a0v = *(const float2*)&lds_a[m * 260 + k + kh];          // 8B aligned
        float2 a1v = *(const float2*)&lds_a[(m + 16) * 260 + k + kh];
        v2f a0; a0.x = a0v.x; a0.y = a0v.y;
        v2f a1; a1.x = a1v.x; a1.y = a1v.y;
        acc0 = __builtin_amdgcn_wmma_f32_16x16x4_f32(
            false, a0, false, b, (short)0, acc0, false, false);
        acc1 = __builtin_amdgcn_wmma_f32_16x16x4_f32(
            false, a1, false, b, (short)0, acc1, false, false);
    }

    const float bv = bias ? bias[col] : 0.0f;
    const int   rb = row0 + ((lane >> 4) << 3);
    #pragma unroll
    for (int i = 0; i < 8; ++i)
        C[(size_t)(rb + i) * HDIM + col] = acc0[i] + bv;
    #pragma unroll
    for (int i = 0; i < 8; ++i)
        C[(size_t)(rb + 16 + i) * HDIM + col] = acc1[i] + bv;
}

// ---------------------------------------------------------------------------
__global__ __launch_bounds__(256) void fill_zero(float* __restrict__ p, int n)
{
    int i = blockIdx.x * 256 + threadIdx.x;
    if (i < n) p[i] = 0.0f;
}

__global__ __launch_bounds__(256) void deg_count(
    const int* __restrict__ tgt, int E, float* __restrict__ deg)
{
    int e = blockIdx.x * 256 + threadIdx.x;
    if (e < E) unsafeAtomicAdd(&deg[tgt[e]], 1.0f);
}

__global__ __launch_bounds__(256) void deg_to_dinv(float* __restrict__ deg, int n)
{
    int i = blockIdx.x * 256 + threadIdx.x;
    if (i < n) deg[i] = rsqrtf(deg[i] + 1.0f);   // self-loop: deg + 1
}

// one wave per edge: agg[tgt,:] += h[src,:] * dinv[src]*dinv[tgt]  (256 cols)
__global__ __launch_bounds__(256) void scatter_gcn(
    const float* __restrict__ h, const int* __restrict__ src,
    const int* __restrict__ tgt, const float* __restrict__ dinv,
    float* __restrict__ agg, int E)
{
    int e    = blockIdx.x * 8 + (threadIdx.x >> 5);
    int lane = threadIdx.x & 31;
    if (e >= E) return;
    int   s    = src[e];
    int   t    = tgt[e];
    float norm = dinv[s] * dinv[t];
    const float4* hp = (const float4*)(h + (size_t)s * HDIM);
    float*        ap = agg + (size_t)t * HDIM;
    #pragma unroll
    for (int j = 0; j < 2; ++j) {
        float4 v = hp[lane * 2 + j];
        int    c = (lane * 2 + j) * 4;
        unsafeAtomicAdd(ap + c + 0, v.x * norm);
        unsafeAtomicAdd(ap + c + 1, v.y * norm);
        unsafeAtomicAdd(ap + c + 2, v.z * norm);
        unsafeAtomicAdd(ap + c + 3, v.w * norm);
    }
}

// one wave per row: x = relu(layernorm(x + (agg + h*dinv^2 + gb)) * lg + lb)
__global__ __launch_bounds__(256) void post_conv(
    float* __restrict__ x, const float* __restrict__ h,
    const float* __restrict__ agg, const float* __restrict__ dinv,
    const float* __restrict__ gb, const float* __restrict__ lg,
    const float* __restrict__ lb, int n)
{
    int row  = blockIdx.x * 8 + (threadIdx.x >> 5);
    int lane = threadIdx.x & 31;
    if (row >= n) return;
    float d2 = dinv[row]; d2 *= d2;
    size_t base = (size_t)row * HDIM;

    float v[8];
    float s = 0.0f;
    #pragma unroll
    for (int j = 0; j < 8; ++j) {
        int   c = lane + j * 32;
        float t = x[base + c] + agg[base + c] + h[base + c] * d2 + gb[c];
        v[j] = t; s += t;
    }
    #pragma unroll
    for (int off = 16; off > 0; off >>= 1) s += __shfl_xor(s, off, 32);
    float mean = s * (1.0f / 256.0f);
    float var = 0.0f;
    #pragma unroll
    for (int j = 0; j < 8; ++j) { float d = v[j] - mean; var += d * d; }
    #pragma unroll
    for (int off = 16; off > 0; off >>= 1) var += __shfl_xor(var, off, 32);
    float rstd = rsqrtf(var * (1.0f / 256.0f) + 1e-5f);
    #pragma unroll
    for (int j = 0; j < 8; ++j) {
        int   c = lane + j * 32;
        float y = (v[j] - mean) * rstd * lg[c] + lb[c];
        x[base + c] = fmaxf(y, 0.0f);
    }
}

// ---------------------------------------------------------------------------
namespace {

struct GnnParams { const float *Win, *bin, *gW, *gb, *lg, *lb, *Wout, *bout; };

inline void run_gemm(const float* A, const float* W, const float* bias,
                     float* C, int K, hipStream_t s)
{
    gemm_bias_wmma<<<dim3(NNODES / 32), 256, 0, s>>>(A, W, bias, C, K);
}

inline void run_gnn(const float* xin, const GnnParams& p, const int* ei, int E,
                    float* xwork, float* hbuf, float* aggbuf, float* dinv,
                    float* outbuf, hipStream_t s)
{
    const int    N  = NNODES;
    const size_t NH = (size_t)N * HDIM;

    run_gemm(xin, p.Win, p.bin, xwork, HDIM, s);           // x = xin@Win + bin

    fill_zero<<<(N + 255) / 256, 256, 0, s>>>(dinv, N);
    deg_count<<<(E + 255) / 256, 256, 0, s>>>(ei + E, E, dinv);   // tgt row
    deg_to_dinv<<<(N + 255) / 256, 256, 0, s>>>(dinv, N);

    for (int l = 0; l < 3; ++l) {
        run_gemm(xwork, p.gW + (size_t)l * HDIM * HDIM, nullptr, hbuf, HDIM, s);
        fill_zero<<<(int)((NH + 255) / 256), 256, 0, s>>>(aggbuf, (int)NH);
        scatter_gcn<<<(E + 7) / 8, 256, 0, s>>>(hbuf, ei, ei + E, dinv, aggbuf, E);
        post_conv<<<(N + 7) / 8, 256, 0, s>>>(xwork, hbuf, aggbuf, dinv,
                                              p.gb + (size_t)l * HDIM,
                                              p.lg, p.lb, N);
    }
    run_gemm(xwork, p.Wout, p.bout, outbuf, HDIM, s);
}

} // namespace

extern "C" void kernel_launch(void* const* d_in, const int* in_sizes, int n_in,
                              void* d_out, int out_size, void* d_ws, size_t ws_size,
                              hipStream_t stream)
{
    (void)in_sizes; (void)n_in; (void)out_size; (void)ws_size;
    const int    N  = NNODES;
    const int    E1 = 640000, E2 = 320000;
    const size_t NH = (size_t)N * HDIM;

    const float* x_stock  = (const float*)d_in[0];
    // d_in[1] x_sector, d_in[4] Wp_sector, d_in[5] bp_sector: dead in reference
    const float* Wp_stock = (const float*)d_in[2];
    const float* bp_stock = (const float*)d_in[3];

    GnnParams pc { (const float*)d_in[6],  (const float*)d_in[7],
                   (const float*)d_in[8],  (const float*)d_in[9],
                   (const float*)d_in[10], (const float*)d_in[11],
                   (const float*)d_in[12], (const float*)d_in[13] };
    GnnParams pb { (const float*)d_in[14], (const float*)d_in[15],
                   (const float*)d_in[16], (const float*)d_in[17],
                   (const float*)d_in[18], (const float*)d_in[19],
                   (const float*)d_in[20], (const float*)d_in[21] };

    const float* Wo_stock  = (const float*)d_in[22];
    const float* bo_stock  = (const float*)d_in[23];
    const float* Wo_sector = (const float*)d_in[24];
    const float* bo_sector = (const float*)d_in[25];
    const int*   ei_corr    = (const int*)d_in[26];
    const int*   ei_belongs = (const int*)d_in[27];

    float* ws   = (float*)d_ws;
    float* bx   = ws;             // N*H
    float* bh   = ws + NH;        // N*H
    float* bagg = ws + 2 * NH;    // N*H
    float* bhs  = ws + 3 * NH;    // N*H (h_stock result)
    float* dinv = ws + 4 * NH;    // N

    // 1) stock input projection (K=64)
    run_gemm(x_stock, Wp_stock, bp_stock, bx, 64, stream);

    // 2) corr GNN on h_stock: xin=bx, xwork=bh, h=bagg, agg=bx(reuse), out=bhs
    run_gnn(bx, pc, ei_corr, E1, bh, bagg, bx, dinv, bhs, stream);

    // 3) belongs GNN with input = corr output: xwork=bx, h=bagg, agg=bh, out=bh
    run_gnn(bhs, pb, ei_belongs, E2, bx, bagg, bh, dinv, bh, stream);

    // 4) output projections
    run_gemm(bhs, Wo_stock,  bo_stock,  (float*)d_out,      HDIM, stream);
    run_gemm(bh,  Wo_sector, bo_sector, (float*)d_out + NH, HDIM, stream);
}